// MinimalQuantumLayer_7851200217705
// MI455X (gfx1250) — compile-verified
//
#include <hip/hip_runtime.h>

typedef float v2f __attribute__((ext_vector_type(2)));
typedef float v8f __attribute__((ext_vector_type(8)));

#define DIM 16

// ---------------------------------------------------------------------------
// Pre-kernel: build the per-layer 16x16 complex unitary
//   W_l = P_cnot_ring * (R_{l,0} on bit8) (R_{l,1} on bit4) (R_{l,2} on bit2) (R_{l,3} on bit1)
// Stored as float2 (re, im), row-major: Wt[l*256 + i*16 + j].
// 768 entries total (6 KB) -> redundant work is negligible.
// ---------------------------------------------------------------------------
__global__ void build_w_kernel(const float* __restrict__ w, float2* __restrict__ Wt) {
    int t = blockIdx.x * blockDim.x + threadIdx.x;
    if (t >= 3 * DIM * DIM) return;
    int l = t >> 8;
    int i = (t >> 4) & 15;
    int j = t & 15;

    // Composed CNOT-ring permutation: p(i) = perm0[perm1[perm2[perm3[i]]]]
    int pi = i;
    #pragma unroll
    for (int q = 3; q >= 0; --q) {
        int tq   = (q + 1) & 3;
        int ctrl = (pi >> (3 - q)) & 1;
        pi = ctrl ? (pi ^ (1 << (3 - tq))) : pi;
    }

    // M[p(i)][j] = prod_q R_q[bit_{3-q}(p(i))][bit_{3-q}(j)]
    float re = 1.0f, im = 0.0f;
    #pragma unroll
    for (int q = 0; q < 4; ++q) {
        float phi   = w[(l * 4 + q) * 3 + 0];
        float theta = w[(l * 4 + q) * 3 + 1];
        float omega = w[(l * 4 + q) * 3 + 2];
        float ct = cosf(0.5f * theta);
        float st = sinf(0.5f * theta);
        int rq = (pi >> (3 - q)) & 1;
        int cq = (j  >> (3 - q)) & 1;
        // R = [[ e^{-i(phi+omega)/2} ct , -e^{ i(phi-omega)/2} st ],
        //      [ e^{-i(phi-omega)/2} st ,  e^{ i(phi+omega)/2} ct ]]
        float ang, mag;
        if (rq == 0 && cq == 0)      { ang = -0.5f * (phi + omega); mag =  ct; }
        else if (rq == 0 && cq == 1) { ang =  0.5f * (phi - omega); mag = -st; }
        else if (rq == 1 && cq == 0) { ang = -0.5f * (phi - omega); mag =  st; }
        else                         { ang =  0.5f * (phi + omega); mag =  ct; }
        float er = mag * cosf(ang);
        float ei = mag * sinf(ang);
        float nr = re * er - im * ei;
        float ni = re * ei + im * er;
        re = nr; im = ni;
    }
    Wt[t] = make_float2(re, im);
}

// ---------------------------------------------------------------------------
// Helpers
// ---------------------------------------------------------------------------
__device__ __forceinline__ float swapx16(float x) {
    // ds_swizzle_b32 group-of-32: and=0x1f, or=0, xor=0x10  -> lane L <-> L^16
    return __int_as_float(__builtin_amdgcn_ds_swizzle(__float_as_int(x), 0x401f));
}

__device__ __forceinline__ v8f wmma4(v2f a, v2f b, v8f c) {
    // D = A(16x4) * B(4x16) + C  in f32
    return __builtin_amdgcn_wmma_f32_16x16x4_f32(false, a, false, b, (short)0, c,
                                                 false, false);
}

// ---------------------------------------------------------------------------
// Main kernel: one wave32 = 16 items. State = two 16x16 f32 matrices (Re/Im)
// held in the WMMA C/D layout (VGPR v: lanes 0-15 -> row v, lanes 16-31 ->
// row v+8; column = lane&15 = item within tile).
// ---------------------------------------------------------------------------
__global__ __launch_bounds__(256) void quantum_kernel(
    const float* __restrict__ x, const float2* __restrict__ Wt,
    float* __restrict__ out, int nItems) {

    const int lane   = threadIdx.x & 31;
    const int waveId = blockIdx.x * (blockDim.x >> 5) + (threadIdx.x >> 5);
    const int base   = waveId << 4;
    if (base >= nItems) return;                 // uniform per wave
    const bool lo    = lane < 16;
    const int  item  = base + (lane & 15);

    // ---- gather 4 corner inputs (SAME-pad 8x8 windows, stride 4) ----
    const int b   = item >> 8;
    const int rem = item & 255;
    const int r0  = ((rem >> 4) << 2) - 2;      // 4*i - 2
    const int c0  = ((rem & 15) << 2) - 2;      // 4*j - 2
    const int dy[4] = {0, 0, 7, 7};
    const int dx[4] = {0, 7, 0, 7};

    float cq[4], sq[4];
    #pragma unroll
    for (int q = 0; q < 4; ++q) {
        int rr = r0 + dy[q], cc = c0 + dx[q];
        bool ok = ((unsigned)rr < 64u) & ((unsigned)cc < 64u);
        int rrc = ok ? rr : 0, ccc = ok ? cc : 0;
        float v = x[(b * 64 + rrc) * 64 + ccc];
        v = ok ? v : 0.0f;
        // ang = atan(v)/2 ; cos(atan v) = 1/sqrt(1+v^2); half-angle identities
        float inv = 1.0f / sqrtf(1.0f + v * v);
        cq[q] = sqrtf(0.5f * (1.0f + inv));
        sq[q] = copysignf(sqrtf(fmaxf(0.0f, 0.5f * (1.0f - inv))), v);
    }
    const float s0n = lo ? -sq[0] : sq[0];      // q0 gate sign per lane-half

    // ---- init state |0000> ----
    float sre[8], sim[8];
    #pragma unroll
    for (int v = 0; v < 8; ++v) { sre[v] = 0.0f; sim[v] = 0.0f; }
    sre[0] = lo ? 1.0f : 0.0f;

    const int m  = lane & 15;                   // A-matrix row for this lane
    const int ko = lo ? 0 : 2;                  // K sub-column pair per half

    #pragma unroll
    for (int l = 0; l < 3; ++l) {
        // ===== data-reuploading RY gates (same angles each layer) =====
        // qubit 0 -> state bit 8: partner lives in the other lane half
        #pragma unroll
        for (int v = 0; v < 8; ++v) {
            float tr = swapx16(sre[v]);
            float ti = swapx16(sim[v]);
            sre[v] = cq[0] * sre[v] + s0n * tr;
            sim[v] = cq[0] * sim[v] + s0n * ti;
        }
        // qubit 1 -> bit 4: VGPR pair (v, v+4)
        #pragma unroll
        for (int v = 0; v < 4; ++v) {
            float ar = sre[v], br = sre[v + 4];
            float ai = sim[v], bi = sim[v + 4];
            sre[v]     = cq[1] * ar - sq[1] * br;
            sre[v + 4] = sq[1] * ar + cq[1] * br;
            sim[v]     = cq[1] * ai - sq[1] * bi;
            sim[v + 4] = sq[1] * ai + cq[1] * bi;
        }
        // qubit 2 -> bit 2: VGPR pair (v, v+2)
        #pragma unroll
        for (int g = 0; g < 4; ++g) {
            int v = (g & 1) + ((g >> 1) << 2);  // 0,1,4,5
            float ar = sre[v], br = sre[v + 2];
            float ai = sim[v], bi = sim[v + 2];
            sre[v]     = cq[2] * ar - sq[2] * br;
            sre[v + 2] = sq[2] * ar + cq[2] * br;
            sim[v]     = cq[2] * ai - sq[2] * bi;
            sim[v + 2] = sq[2] * ai + cq[2] * bi;
        }
        // qubit 3 -> bit 1: VGPR pair (v, v+1)
        #pragma unroll
        for (int g = 0; g < 4; ++g) {
            int v = g << 1;                     // 0,2,4,6
            float ar = sre[v], br = sre[v + 1];
            float ai = sim[v], bi = sim[v + 1];
            sre[v]     = cq[3] * ar - sq[3] * br;
            sre[v + 1] = sq[3] * ar + cq[3] * br;
            sim[v]     = cq[3] * ai - sq[3] * bi;
            sim[v + 1] = sq[3] * ai + cq[3] * bi;
        }

        // ===== repack C/D layout -> B layout (rows 4k..4k+3 per chunk) =====
        v2f bre[4], bim[4], bneg[4];
        {
            float wr0 = swapx16(sre[0]), wr1 = swapx16(sre[1]);
            float wr2 = swapx16(sre[2]), wr3 = swapx16(sre[3]);
            float wr4 = swapx16(sre[4]), wr5 = swapx16(sre[5]);
            float wr6 = swapx16(sre[6]), wr7 = swapx16(sre[7]);
            bre[0] = (v2f){lo ? sre[0] : wr2, lo ? sre[1] : wr3};   // rows 0/2, 1/3
            bre[1] = (v2f){lo ? sre[4] : wr6, lo ? sre[5] : wr7};   // rows 4/6, 5/7
            bre[2] = (v2f){lo ? wr0 : sre[2], lo ? wr1 : sre[3]};   // rows 8/10, 9/11
            bre[3] = (v2f){lo ? wr4 : sre[6], lo ? wr5 : sre[7]};   // rows 12/14, 13/15
            float wi0 = swapx16(sim[0]), wi1 = swapx16(sim[1]);
            float wi2 = swapx16(sim[2]), wi3 = swapx16(sim[3]);
            float wi4 = swapx16(sim[4]), wi5 = swapx16(sim[5]);
            float wi6 = swapx16(sim[6]), wi7 = swapx16(sim[7]);
            bim[0] = (v2f){lo ? sim[0] : wi2, lo ? sim[1] : wi3};
            bim[1] = (v2f){lo ? sim[4] : wi6, lo ? sim[5] : wi7};
            bim[2] = (v2f){lo ? wi0 : sim[2], lo ? wi1 : sim[3]};
            bim[3] = (v2f){lo ? wi4 : sim[6], lo ? wi5 : sim[7]};
            #pragma unroll
            for (int kc = 0; kc < 4; ++kc)
                bneg[kc] = (v2f){-bim[kc].x, -bim[kc].y};
        }

        // ===== load W_l in A layout (row m, cols 4kc+ko, 4kc+ko+1) =====
        v2f wre[4], wim[4];
        const float2* Wl = Wt + l * 256 + m * 16;
        #pragma unroll
        for (int kc = 0; kc < 4; ++kc) {
            float2 e0 = Wl[4 * kc + ko];
            float2 e1 = Wl[4 * kc + ko + 1];
            wre[kc] = (v2f){e0.x, e1.x};
            wim[kc] = (v2f){e0.y, e1.y};
        }

        // ===== S' = W_l * S  (complex) via 16x v_wmma_f32_16x16x4_f32 =====
        v8f dre = {}, dim = {};
        #pragma unroll
        for (int kc = 0; kc < 4; ++kc) dre = wmma4(wre[kc], bre[kc], dre);
        #pragma unroll
        for (int kc = 0; kc < 4; ++kc) dre = wmma4(wim[kc], bneg[kc], dre);
        #pragma unroll
        for (int kc = 0; kc < 4; ++kc) dim = wmma4(wre[kc], bim[kc], dim);
        #pragma unroll
        for (int kc = 0; kc < 4; ++kc) dim = wmma4(wim[kc], bre[kc], dim);
        #pragma unroll
        for (int v = 0; v < 8; ++v) { sre[v] = dre[v]; sim[v] = dim[v]; }
    }

    // ===== Z expectations: out[w] = sum_M (1 - 2*bit_{3-w}(M)) * |amp_M|^2 =====
    float p[8];
    #pragma unroll
    for (int v = 0; v < 8; ++v) p[v] = sre[v] * sre[v] + sim[v] * sim[v];

    float e0 = (p[0] + p[1] + p[2] + p[3]) + (p[4] + p[5] + p[6] + p[7]);
    e0 = lo ? e0 : -e0;                                   // bit 8 = lane half
    float e1 = (p[0] + p[1] + p[2] + p[3]) - (p[4] + p[5] + p[6] + p[7]);
    float e2 = (p[0] + p[1] + p[4] + p[5]) - (p[2] + p[3] + p[6] + p[7]);
    float e3 = (p[0] + p[2] + p[4] + p[6]) - (p[1] + p[3] + p[5] + p[7]);

    e0 += swapx16(e0);
    e1 += swapx16(e1);
    e2 += swapx16(e2);
    e3 += swapx16(e3);

    if (lo && item < nItems) {
        float4 o = make_float4(e0, e1, e2, e3);
        *reinterpret_cast<float4*>(out + (size_t)item * 4) = o;  // b128 store
    }
}

// ---------------------------------------------------------------------------
extern "C" void kernel_launch(void* const* d_in, const int* in_sizes, int n_in,
                              void* d_out, int out_size, void* d_ws, size_t ws_size,
                              hipStream_t stream) {
    const float* x  = (const float*)d_in[0];   // (4096,64,64,1) f32
    const float* qw = (const float*)d_in[1];   // (3,4,3) f32
    float*  out = (float*)d_out;               // (4096,16,16,4) f32
    float2* Wt  = (float2*)d_ws;               // 768 complex entries (6 KB)

    build_w_kernel<<<3, 256, 0, stream>>>(qw, Wt);

    const int nItems = out_size / 4;           // 1,048,576
    const int tiles  = (nItems + 15) / 16;     // one wave32 per 16 items
    const int blocks = (tiles + 7) / 8;        // 8 waves per 256-thread block
    quantum_kernel<<<blocks, 256, 0, stream>>>(x, Wt, out, nItems);
}